// DotLayer_10239202034235
// MI455X (gfx1250) — compile-verified
//
#include <hip/hip_runtime.h>

// Cosine similarity per row: a,b are [64, 2048, 512] fp32 -> out [64, 2048] fp32.
// Pure HBM-streaming kernel (512 MiB read once, AI ~= 0.375 FLOP/B):
//   - one wave32 per 512-float row, lane l owns float4 at element 4*l of each
//     128-float chunk -> fully coalesced 512B global_load_b128 bursts
//   - nontemporal loads (TH=NT) so streamed-once data doesn't thrash caches
//   - 5-step wave32 butterfly reduction of (a.b, a.a, b.b)
//   - lane 0 combines with rsqrt and stores.

typedef float f32x4 __attribute__((ext_vector_type(4)));

#ifndef COSROW_EPS
#define COSROW_EPS 1e-12f
#endif

__global__ __launch_bounds__(256) void cosine_rows_kernel(
    const float* __restrict__ a,
    const float* __restrict__ b,
    float* __restrict__ out,
    int nrows) {
  const int lane   = threadIdx.x & 31;
  const int waveId = threadIdx.x >> 5;
  const int row    = blockIdx.x * (blockDim.x >> 5) + waveId;
  if (row >= nrows) return;

  const size_t rowBase = (size_t)row * 512u;
  const f32x4* __restrict__ arow =
      reinterpret_cast<const f32x4*>(a + rowBase) + lane;
  const f32x4* __restrict__ brow =
      reinterpret_cast<const f32x4*>(b + rowBase) + lane;

  // Issue all 8 B128 loads up-front (8 x 512B in flight per wave) so the
  // compiler emits the loads as one clause and a single loadcnt wait.
  f32x4 av[4], bv[4];
#pragma unroll
  for (int c = 0; c < 4; ++c)
    av[c] = __builtin_nontemporal_load(arow + c * 32);  // chunk stride = 128 floats
#pragma unroll
  for (int c = 0; c < 4; ++c)
    bv[c] = __builtin_nontemporal_load(brow + c * 32);

  float dab = 0.0f, saa = 0.0f, sbb = 0.0f;
#pragma unroll
  for (int c = 0; c < 4; ++c) {
#pragma unroll
    for (int k = 0; k < 4; ++k) {
      const float x = av[c][k];
      const float y = bv[c][k];
      dab = fmaf(x, y, dab);
      saa = fmaf(x, x, saa);
      sbb = fmaf(y, y, sbb);
    }
  }

  // wave32 butterfly reduction (5 steps, NOT 6 — gfx1250 is wave32-only).
#pragma unroll
  for (int m = 16; m >= 1; m >>= 1) {
    dab += __shfl_xor(dab, m, 32);
    saa += __shfl_xor(saa, m, 32);
    sbb += __shfl_xor(sbb, m, 32);
  }

  if (lane == 0) {
    // sum(an*bn) == dot(a,b) * rsqrt(max(|a|^2,eps)) * rsqrt(max(|b|^2,eps))
    const float r = dab * rsqrtf(fmaxf(saa, COSROW_EPS))
                        * rsqrtf(fmaxf(sbb, COSROW_EPS));
    out[row] = r;
  }
}

extern "C" void kernel_launch(void* const* d_in, const int* in_sizes, int n_in,
                              void* d_out, int out_size, void* d_ws, size_t ws_size,
                              hipStream_t stream) {
  (void)n_in; (void)d_ws; (void)ws_size;
  const float* a = (const float*)d_in[0];
  const float* b = (const float*)d_in[1];
  float* out = (float*)d_out;

  const int nrows = out_size;               // 64 * 2048 = 131072 rows of 512
  (void)in_sizes;

  const int threads = 256;                  // 8 wave32's -> 8 rows per block
  const int rowsPerBlock = threads / 32;
  const int blocks = (nrows + rowsPerBlock - 1) / rowsPerBlock;

  cosine_rows_kernel<<<blocks, threads, 0, stream>>>(a, b, out, nrows);
}